// GCNSuperToken_515396075767
// MI455X (gfx1250) — compile-verified
//
#include <hip/hip_runtime.h>

typedef __attribute__((ext_vector_type(2))) float v2f;
typedef __attribute__((ext_vector_type(8))) float v8f;

#define DFEAT 128

// ---------------------------------------------------------------------------
// Degree accumulation: one thread per edge, float atomics.
// ---------------------------------------------------------------------------
__global__ void gcn_deg_kernel(const int* __restrict__ src, const int* __restrict__ dst,
                               float* __restrict__ deg_out, float* __restrict__ deg_in, int E) {
  int e = blockIdx.x * blockDim.x + threadIdx.x;
  if (e < E) {
    atomicAdd(&deg_out[src[e]], 1.0f);
    atomicAdd(&deg_in[dst[e]], 1.0f);
  }
}

// In-place convert degrees -> clip(deg,1)^-1/2
__global__ void gcn_norm_kernel(float* __restrict__ a, float* __restrict__ b, int N) {
  int i = blockIdx.x * blockDim.x + threadIdx.x;
  if (i < N) {
    a[i] = rsqrtf(fmaxf(a[i], 1.0f));
    b[i] = rsqrtf(fmaxf(b[i], 1.0f));
  }
}

// ---------------------------------------------------------------------------
// Gather + scale + scatter-add: one wave32 per edge, one float4 per lane.
// ---------------------------------------------------------------------------
__global__ __launch_bounds__(256) void gcn_scatter_kernel(
    const float* __restrict__ feat, const float* __restrict__ norm_src,
    const int* __restrict__ src, const int* __restrict__ dst,
    float* __restrict__ agg, int E) {
  const int lane = threadIdx.x & 31;
  const int waveId = (blockIdx.x * blockDim.x + threadIdx.x) >> 5;
  const int nWaves = (gridDim.x * blockDim.x) >> 5;

  for (int e = waveId; e < E; e += nWaves) {
    int s = src[e];
    int d = dst[e];
    // Prefetch the feature row for this wave slot's next edge (global_prefetch_b8).
    int e2 = e + nWaves;
    if (e2 < E) {
      int s2 = src[e2];
      __builtin_prefetch((const void*)(feat + (size_t)s2 * DFEAT + lane * 4), 0, 0);
    }
    float w = norm_src[s];
    float4 v = ((const float4*)(feat + (size_t)s * DFEAT))[lane];
    float* o = agg + (size_t)d * DFEAT + lane * 4;
    atomicAdd(o + 0, v.x * w);
    atomicAdd(o + 1, v.y * w);
    atomicAdd(o + 2, v.z * w);
    atomicAdd(o + 3, v.w * w);
  }
}

// ---------------------------------------------------------------------------
// Fused  out = relu( diag(norm_dst) * A @ W + b )  with exact-fp32 WMMA.
// Block = 256 threads (8 waves). Block computes a 64x128 output band.
// Wave w owns column tile w (16 cols) and 4 row tiles (4 x v8f accumulators).
// K reduction: 32 chained V_WMMA_F32_16X16X4_F32 per tile.
// ---------------------------------------------------------------------------
__global__ __launch_bounds__(256) void gcn_gemm_kernel(
    const float* __restrict__ A, const float* __restrict__ norm_dst,
    const float* __restrict__ W, const float* __restrict__ bias,
    float* __restrict__ out, int N) {
  __shared__ float sA[64 * DFEAT];   // 32 KB: A band, pre-scaled by norm_dst
  __shared__ float sW[32 * DFEAT];   // 16 KB: W k-chunk
  __shared__ float sB[DFEAT];        // bias

  const int tid = threadIdx.x;
  const int wave = tid >> 5;
  const int lane = tid & 31;
  const int row0 = blockIdx.x * 64;
  const int col0 = wave * 16;
  const int nl = lane & 15;          // column/row-within-tile index
  const int koff = (lane >> 4) << 1; // lanes 0-15 -> K{0,1}, lanes 16-31 -> K{2,3}

  if (tid < DFEAT) sB[tid] = bias[tid];

  // Load + scale A band (64 rows x 128 cols): 2048 float4, 8 per thread.
#pragma unroll
  for (int it = 0; it < 8; ++it) {
    int idx4 = tid + it * 256;   // float4 index in band
    int r = idx4 >> 5;           // local row (32 float4 per row)
    int c4 = idx4 & 31;
    int row = row0 + r;
    float4 v = {0.0f, 0.0f, 0.0f, 0.0f};
    if (row < N) {
      v = ((const float4*)(A + (size_t)row * DFEAT))[c4];
      float sc = norm_dst[row];
      v.x *= sc; v.y *= sc; v.z *= sc; v.w *= sc;
    }
    ((float4*)sA)[idx4] = v;
  }

  v8f acc0 = {}, acc1 = {}, acc2 = {}, acc3 = {};

  for (int kc = 0; kc < DFEAT; kc += 32) {
    __syncthreads();  // previous chunk fully consumed (and A band ready on iter 0)
    // Stream W rows [kc, kc+32) into LDS: 1024 float4, 4 per thread.
#pragma unroll
    for (int it = 0; it < 4; ++it) {
      int idx4 = tid + it * 256;
      ((float4*)sW)[idx4] = ((const float4*)(W + (size_t)kc * DFEAT))[idx4];
    }
    __syncthreads();

#pragma unroll
    for (int ks = 0; ks < 8; ++ks) {
      const int kl = ks * 4 + koff;        // local K row pair base in sW
      const int kg = kc + kl;              // global K index into sA rows

      v2f bfrag;                           // B is KxN: W[k][col]
      bfrag[0] = sW[kl * DFEAT + col0 + nl];
      bfrag[1] = sW[(kl + 1) * DFEAT + col0 + nl];

      v2f a0, a1, a2, a3;                  // A is MxK, M = tile*16 + nl
      a0[0] = sA[(0 * 16 + nl) * DFEAT + kg]; a0[1] = sA[(0 * 16 + nl) * DFEAT + kg + 1];
      a1[0] = sA[(1 * 16 + nl) * DFEAT + kg]; a1[1] = sA[(1 * 16 + nl) * DFEAT + kg + 1];
      a2[0] = sA[(2 * 16 + nl) * DFEAT + kg]; a2[1] = sA[(2 * 16 + nl) * DFEAT + kg + 1];
      a3[0] = sA[(3 * 16 + nl) * DFEAT + kg]; a3[1] = sA[(3 * 16 + nl) * DFEAT + kg + 1];

      acc0 = __builtin_amdgcn_wmma_f32_16x16x4_f32(false, a0, false, bfrag, (short)0, acc0, false, false);
      acc1 = __builtin_amdgcn_wmma_f32_16x16x4_f32(false, a1, false, bfrag, (short)0, acc1, false, false);
      acc2 = __builtin_amdgcn_wmma_f32_16x16x4_f32(false, a2, false, bfrag, (short)0, acc2, false, false);
      acc3 = __builtin_amdgcn_wmma_f32_16x16x4_f32(false, a3, false, bfrag, (short)0, acc3, false, false);
    }
  }

  // C/D layout: VGPR j -> M=j (lanes 0-15) or M=j+8 (lanes 16-31); N = lane&15.
  const int c = col0 + nl;
  const float bval = sB[c];
  const int mhalf = (lane >> 4) << 3;
#pragma unroll
  for (int j = 0; j < 8; ++j) {
    int ml = j + mhalf;
    int r;
    r = row0 + 0 * 16 + ml; if (r < N) out[(size_t)r * DFEAT + c] = fmaxf(acc0[j] + bval, 0.0f);
    r = row0 + 1 * 16 + ml; if (r < N) out[(size_t)r * DFEAT + c] = fmaxf(acc1[j] + bval, 0.0f);
    r = row0 + 2 * 16 + ml; if (r < N) out[(size_t)r * DFEAT + c] = fmaxf(acc2[j] + bval, 0.0f);
    r = row0 + 3 * 16 + ml; if (r < N) out[(size_t)r * DFEAT + c] = fmaxf(acc3[j] + bval, 0.0f);
  }
}

// ---------------------------------------------------------------------------
extern "C" void kernel_launch(void* const* d_in, const int* in_sizes, int n_in,
                              void* d_out, int out_size, void* d_ws, size_t ws_size,
                              hipStream_t stream) {
  const float* feat = (const float*)d_in[0];
  const int* src    = (const int*)d_in[1];
  const int* dst    = (const int*)d_in[2];
  const float* W1   = (const float*)d_in[3];
  const float* b1   = (const float*)d_in[4];
  const float* W2   = (const float*)d_in[5];
  const float* b2   = (const float*)d_in[6];
  float* out        = (float*)d_out;

  const int N = in_sizes[0] / DFEAT;
  const int E = in_sizes[1];

  // Workspace layout: [norm_src N][norm_dst N][agg N*128][h1 N*128]
  float* norm_src = (float*)d_ws;
  float* norm_dst = norm_src + N;
  float* agg      = norm_dst + N;
  float* h1       = agg + (size_t)N * DFEAT;

  // Degrees -> norms
  hipMemsetAsync(norm_src, 0, sizeof(float) * 2 * (size_t)N, stream);
  gcn_deg_kernel<<<(E + 255) / 256, 256, 0, stream>>>(src, dst, norm_src, norm_dst, E);
  gcn_norm_kernel<<<(N + 255) / 256, 256, 0, stream>>>(norm_src, norm_dst, N);

  const int scatterBlocks = 4096;            // 32768 persistent waves, grid-stride
  const int gemmBlocks = (N + 63) / 64;

  // Layer 1
  hipMemsetAsync(agg, 0, sizeof(float) * (size_t)N * DFEAT, stream);
  gcn_scatter_kernel<<<scatterBlocks, 256, 0, stream>>>(feat, norm_src, src, dst, agg, E);
  gcn_gemm_kernel<<<gemmBlocks, 256, 0, stream>>>(agg, norm_dst, W1, b1, h1, N);

  // Layer 2
  hipMemsetAsync(agg, 0, sizeof(float) * (size_t)N * DFEAT, stream);
  gcn_scatter_kernel<<<scatterBlocks, 256, 0, stream>>>(h1, norm_src, src, dst, agg, E);
  gcn_gemm_kernel<<<gemmBlocks, 256, 0, stream>>>(agg, norm_dst, W2, b2, out, N);
}